// Net_28991029248568
// MI455X (gfx1250) — compile-verified
//
#include <hip/hip_runtime.h>

// ---------------------------------------------------------------------------
// GraphConv x2 + mean-pool + linear head for MI455X (gfx1250, wave32).
//
// Trick: agg @ W_rel == segment_sum((x @ W_rel)[src]) -> project node features
// FIRST with V_WMMA_F32_16X16X4_F32, then scatter only 16-float rows per edge.
// All scatter targets (6.4 MB per buffer) live comfortably in the 192 MB L2,
// so f32 atomics resolve near-chip. fp32 WMMA keeps full precision.
// ---------------------------------------------------------------------------

typedef float v2f __attribute__((ext_vector_type(2)));
typedef float v8f __attribute__((ext_vector_type(8)));

// D[N,16] = X[N,K] @ W[K,16] for two weight matrices at once (shared A tile).
// A (16x4 f32) layout: VGPR0: lanes0-15 K=k0,   lanes16-31 K=k0+2
//                      VGPR1: lanes0-15 K=k0+1, lanes16-31 K=k0+3
//  -> lane loads contiguous float2 at X[row0+(lane&15)][k0+2*(lane>>4)].
// B (4x16) assumed mirrored: b.x = W[k0+2*hi][n], b.y = W[k0+2*hi+1][n], n=lane&15.
// C/D: VGPR r -> D[r][n] (lanes 0-15), D[r+8][n] (lanes 16-31).
template <int K>
__global__ void project_wmma(const float* __restrict__ X,
                             const float* __restrict__ Wrel,
                             const float* __restrict__ Wroot,
                             const float* __restrict__ bias,
                             float* __restrict__ OutRel,    // [N,16] = X@Wrel
                             float* __restrict__ OutRoot,   // [N,16] = X@Wroot + bias
                             int ntiles) {
  const int wave = (int)((blockIdx.x * blockDim.x + threadIdx.x) >> 5);
  const int lane = (int)(threadIdx.x & 31);
  if (wave >= ntiles) return;              // wave-uniform: EXEC stays all-1s
  const int n  = lane & 15;
  const int hi = lane >> 4;
  const float* xrow = X + (size_t)(wave * 16 + n) * K;   // row for A operand
  v8f crel  = {};
  v8f croot = {};
#pragma unroll
  for (int k0 = 0; k0 < K; k0 += 4) {
    const int ka = k0 + 2 * hi;
    v2f a;
    a.x = xrow[ka];
    a.y = xrow[ka + 1];
    v2f brel, broot;
    brel.x  = Wrel [ka * 16 + n];
    brel.y  = Wrel [(ka + 1) * 16 + n];
    broot.x = Wroot[ka * 16 + n];
    broot.y = Wroot[(ka + 1) * 16 + n];
    crel  = __builtin_amdgcn_wmma_f32_16x16x4_f32(false, a, false, brel,
                                                  (short)0, crel,  false, false);
    croot = __builtin_amdgcn_wmma_f32_16x16x4_f32(false, a, false, broot,
                                                  (short)0, croot, false, false);
  }
  const float bn = bias[n];
  float* orel  = OutRel  + (size_t)wave * 256;
  float* oroot = OutRoot + (size_t)wave * 256;
#pragma unroll
  for (int r = 0; r < 8; ++r) {
    const int rr = r + 8 * hi;
    orel [rr * 16 + n] = crel[r];
    oroot[rr * 16 + n] = croot[r] + bn;
  }
}

// Scalar tail for node counts not divisible by 16 (unused when N%16==0).
__global__ void project_tail(const float* __restrict__ X, int K,
                             const float* __restrict__ Wrel,
                             const float* __restrict__ Wroot,
                             const float* __restrict__ bias,
                             float* __restrict__ OutRel,
                             float* __restrict__ OutRoot,
                             int row0, int nrows) {
  const int t = blockIdx.x * blockDim.x + threadIdx.x;
  const int r = t >> 4, n = t & 15;
  if (r >= nrows) return;
  const int row = row0 + r;
  float arel = 0.f, aroot = 0.f;
  for (int k = 0; k < K; ++k) {
    const float xv = X[(size_t)row * K + k];
    arel  += xv * Wrel [k * 16 + n];
    aroot += xv * Wroot[k * 16 + n];
  }
  OutRel [(size_t)row * 16 + n] = arel;
  OutRoot[(size_t)row * 16 + n] = aroot + bias[n];
}

__global__ void zero_f32(float* __restrict__ p, int n) {
  const int t = blockIdx.x * blockDim.x + threadIdx.x;
  if (t < n) p[t] = 0.0f;
}

// agg[dst[e]][:] += feat[src[e]][:]; 4 threads/edge, float4 payload each.
__global__ void scatter16(const float* __restrict__ feat,
                          const int* __restrict__ srcIdx,
                          const int* __restrict__ dstIdx,
                          float* __restrict__ agg,
                          int nEdges) {
  const int t = blockIdx.x * blockDim.x + threadIdx.x;
  const int e = t >> 2, q = t & 3;
  if (e >= nEdges) return;
  const int s = srcIdx[e];
  const int d = dstIdx[e];
  const float4 v = *(const float4*)(feat + (size_t)s * 16 + q * 4);
  float* o = agg + (size_t)d * 16 + q * 4;
  atomicAdd(o + 0, v.x);
  atomicAdd(o + 1, v.y);
  atomicAdd(o + 2, v.z);
  atomicAdd(o + 3, v.w);
}

// h = relu(agg + r); also re-zeros agg so it can be reused by layer 2.
__global__ void combine_relu(float* __restrict__ agg,
                             const float* __restrict__ r,
                             float* __restrict__ h, int n4) {
  const int t = blockIdx.x * blockDim.x + threadIdx.x;
  if (t >= n4) return;
  const float4 a = ((const float4*)agg)[t];
  const float4 b = ((const float4*)r)[t];
  float4 o;
  o.x = fmaxf(a.x + b.x, 0.f);
  o.y = fmaxf(a.y + b.y, 0.f);
  o.z = fmaxf(a.z + b.z, 0.f);
  o.w = fmaxf(a.w + b.w, 0.f);
  ((float4*)h)[t] = o;
  ((float4*)agg)[t] = make_float4(0.f, 0.f, 0.f, 0.f);
}

// h2 = agg + r (layer-2 output, never materialized) accumulated into per-graph
// sums; one count increment per node.
__global__ void combine_pool(const float* __restrict__ agg,
                             const float* __restrict__ r,
                             const int* __restrict__ batch,
                             float* __restrict__ sums,
                             float* __restrict__ counts,
                             int nNodes) {
  const int t = blockIdx.x * blockDim.x + threadIdx.x;
  const int i = t >> 2, q = t & 3;
  if (i >= nNodes) return;
  const int g = batch[i];
  const float4 a = ((const float4*)(agg + (size_t)i * 16))[q];
  const float4 b = ((const float4*)(r   + (size_t)i * 16))[q];
  float* s = sums + (size_t)g * 16 + q * 4;
  atomicAdd(s + 0, a.x + b.x);
  atomicAdd(s + 1, a.y + b.y);
  atomicAdd(s + 2, a.z + b.z);
  atomicAdd(s + 3, a.w + b.w);
  if (q == 0) atomicAdd(&counts[g], 1.0f);
}

// out[g] = dot(sums[g]/max(cnt,1), Wlin) + blin
__global__ void final_head(const float* __restrict__ sums,
                           const float* __restrict__ counts,
                           const float* __restrict__ Wlin,
                           const float* __restrict__ blin,
                           float* __restrict__ out, int nGraphs) {
  const int g = blockIdx.x * blockDim.x + threadIdx.x;
  if (g >= nGraphs) return;
  const float c = fmaxf(counts[g], 1.0f);
  float acc = 0.f;
#pragma unroll
  for (int j = 0; j < 16; ++j) acc += sums[(size_t)g * 16 + j] * Wlin[j];
  out[g] = acc / c + blin[0];
}

extern "C" void kernel_launch(void* const* d_in, const int* in_sizes, int n_in,
                              void* d_out, int out_size, void* d_ws, size_t ws_size,
                              hipStream_t stream) {
  const float* x       = (const float*)d_in[0];
  const int*   edge    = (const int*)  d_in[1];   // [2, E] (src row, dst row)
  const int*   batch   = (const int*)  d_in[2];
  const float* W1_rel  = (const float*)d_in[3];
  const float* W1_root = (const float*)d_in[4];
  const float* b1      = (const float*)d_in[5];
  const float* W2_rel  = (const float*)d_in[6];
  const float* W2_root = (const float*)d_in[7];
  const float* b2      = (const float*)d_in[8];
  const float* Wlin    = (const float*)d_in[9];
  const float* blin    = (const float*)d_in[10];
  float* out = (float*)d_out;

  const int nNodes  = in_sizes[0] / 64;   // 100000
  const int nEdges  = in_sizes[1] / 2;    // 800000
  const int nGraphs = out_size;           // 64
  const int tiles   = nNodes / 16;        // 6250 (exact for 100000)
  const int tail    = nNodes - tiles * 16;
  const int* srcIdx = edge;
  const int* dstIdx = edge + nEdges;

  // Workspace layout (floats): A | B | H | AGG | SUMS | CNT   (~25.6 MB)
  float* A    = (float*)d_ws;                 // proj-by-W_rel  [N,16]
  float* B    = A   + (size_t)nNodes * 16;    // proj-by-W_root (+bias) [N,16]
  float* H    = B   + (size_t)nNodes * 16;    // h1 [N,16]
  float* AGG  = H   + (size_t)nNodes * 16;    // scatter target [N,16]
  float* SUMS = AGG + (size_t)nNodes * 16;    // [G,16]
  float* CNT  = SUMS + (size_t)nGraphs * 16;  // [G]

  const int TB = 256;
  const int zeroN = nNodes * 16 + nGraphs * 16 + nGraphs;  // AGG+SUMS+CNT (contiguous)
  zero_f32<<<(zeroN + TB - 1) / TB, TB, 0, stream>>>(AGG, zeroN);

  const int projBlocks = (tiles + 7) / 8;  // 8 waves (256 thr) per block
  // ---- Layer 1: project (WMMA, K=64), scatter, relu-combine ----
  project_wmma<64><<<projBlocks, TB, 0, stream>>>(x, W1_rel, W1_root, b1, A, B, tiles);
  if (tail > 0)
    project_tail<<<(tail * 16 + TB - 1) / TB, TB, 0, stream>>>(
        x, 64, W1_rel, W1_root, b1, A, B, tiles * 16, tail);
  scatter16<<<((size_t)nEdges * 4 + TB - 1) / TB, TB, 0, stream>>>(A, srcIdx, dstIdx, AGG, nEdges);
  combine_relu<<<(nNodes * 4 + TB - 1) / TB, TB, 0, stream>>>(AGG, B, H, nNodes * 4);

  // ---- Layer 2: project (WMMA, K=16), scatter, combine+pool ----
  project_wmma<16><<<projBlocks, TB, 0, stream>>>(H, W2_rel, W2_root, b2, A, B, tiles);
  if (tail > 0)
    project_tail<<<(tail * 16 + TB - 1) / TB, TB, 0, stream>>>(
        H, 16, W2_rel, W2_root, b2, A, B, tiles * 16, tail);
  scatter16<<<((size_t)nEdges * 4 + TB - 1) / TB, TB, 0, stream>>>(A, srcIdx, dstIdx, AGG, nEdges);
  combine_pool<<<(nNodes * 4 + TB - 1) / TB, TB, 0, stream>>>(AGG, B, batch, SUMS, CNT, nNodes);

  // ---- Head ----
  final_head<<<1, 64, 0, stream>>>(SUMS, CNT, Wlin, blin, out, nGraphs);
}